// SymmetricSelfPacedLearning_11407433138209
// MI455X (gfx1250) — compile-verified
//
#include <hip/hip_runtime.h>
#include <math.h>

typedef __attribute__((ext_vector_type(2))) float v2f;
typedef __attribute__((ext_vector_type(8))) float v8f;

#define B_N 4096
#define D_N 16384

// ---------------------------------------------------------------------------
// Kernel 1: per-row sum of squares via V_WMMA_F32_16X16X4_F32 with A = ones.
// One block (8 wave32) per row; each lane streams float4 chunks (coalesced
// b128 loads), squares them, and accumulates through the WMMA unit.
// HBM-bound: 268 MB @ 23.3 TB/s ~= 11.5 us.
// Writes difficulty[row] = 0.5*loss + 0.5*sqrt(sumsq) to out[1+row].
// ---------------------------------------------------------------------------
__global__ __launch_bounds__(256) void sspl_rowsumsq(
    const float* __restrict__ loss,
    const float* __restrict__ grads,
    float* __restrict__ out)
{
    const int row  = blockIdx.x;
    const int tid  = threadIdx.x;   // 0..255
    const int lane = tid & 31;
    const int wave = tid >> 5;      // 0..7

    const float* gr = grads + (size_t)row * (size_t)D_N;

    v8f c = {0.f, 0.f, 0.f, 0.f, 0.f, 0.f, 0.f, 0.f};
    v2f a = {1.0f, 1.0f};  // ones A-matrix: D[m][n] = sum_k B[k][n]

    // 16 iterations: each iteration the block covers 256*4 = 1024 elements.
    // Uniform trip count -> EXEC stays all-1s for every WMMA (ISA requirement).
#pragma unroll 4
    for (int it = 0; it < D_N / (256 * 4); ++it) {
        const float4 g = *reinterpret_cast<const float4*>(gr + (it * 256 + tid) * 4);
        v2f b0 = {g.x * g.x, g.y * g.y};
        v2f b1 = {g.z * g.z, g.w * g.w};
        // 8 args: (neg_a, A, neg_b, B, c_mod, C, reuse_a, reuse_b)
        c = __builtin_amdgcn_wmma_f32_16x16x4_f32(false, a, false, b0, (short)0, c, false, false);
        c = __builtin_amdgcn_wmma_f32_16x16x4_f32(false, a, false, b1, (short)0, c, false, false);
    }

    // C layout: lane n (0-15) VGPR0 = C[M=0][n], lane n+16 VGPR0 = C[M=8][n];
    // all M rows are identical column sums, and columns are duplicated across
    // the two half-wave groups -> reduce c[0] over 32 lanes, then halve.
    float p = c[0];
    for (int off = 16; off > 0; off >>= 1)
        p += __shfl_xor(p, off, 32);
    p *= 0.5f;  // undo half-wave duplication

    __shared__ float s[8];
    if (lane == 0) s[wave] = p;
    __syncthreads();
    if (tid == 0) {
        float t = 0.f;
        for (int w = 0; w < 8; ++w) t += s[w];
        const float grad_len = sqrtf(t);
        out[1 + row] = 0.5f * loss[row] + 0.5f * grad_len;
    }
}

// ---------------------------------------------------------------------------
// Kernel 2: stable rank of each difficulty (matches jnp.argsort tie-break),
// weight = w_first - step*rank, deterministic per-block partial of
// loss[j]*weight[j] via LDS tree reduction. 16 blocks x 256 threads.
// ---------------------------------------------------------------------------
__global__ __launch_bounds__(256) void sspl_rank(
    const float* __restrict__ loss,
    const float* __restrict__ diff,
    float* __restrict__ partials,
    float w_first, float step)
{
    __shared__ float sd[B_N];       // 16 KB of 320 KB WGP LDS
    __shared__ float sred[256];

    const int tid = threadIdx.x;
    for (int i = tid; i < B_N; i += 256) sd[i] = diff[i];
    __syncthreads();

    const int j = blockIdx.x * 256 + tid;
    const float dj = sd[j];
    int rank = 0;
#pragma unroll 8
    for (int k = 0; k < B_N; ++k) {
        const float dk = sd[k];
        rank += (int)((dk < dj) || (dk == dj && k < j));
    }
    sred[tid] = loss[j] * (w_first - step * (float)rank);
    __syncthreads();

    for (int s2 = 128; s2 > 0; s2 >>= 1) {
        if (tid < s2) sred[tid] += sred[tid + s2];
        __syncthreads();
    }
    if (tid == 0) partials[blockIdx.x] = sred[0];
}

// ---------------------------------------------------------------------------
// Kernel 3: fixed-order sum of 16 partials -> weighted_loss (deterministic).
// ---------------------------------------------------------------------------
__global__ void sspl_final(const float* __restrict__ partials,
                           float* __restrict__ out)
{
    if (threadIdx.x == 0 && blockIdx.x == 0) {
        float s = 0.f;
        for (int i = 0; i < 16; ++i) s += partials[i];
        out[0] = s * (1.0f / (float)B_N);
    }
}

// ---------------------------------------------------------------------------
extern "C" void kernel_launch(void* const* d_in, const int* in_sizes, int n_in,
                              void* d_out, int out_size, void* d_ws, size_t ws_size,
                              hipStream_t stream)
{
    const float* loss  = (const float*)d_in[0];   // [B] f32
    const float* grads = (const float*)d_in[1];   // [B, D] f32
    float* out = (float*)d_out;                   // [1 + B] f32
    float* partials = (float*)d_ws;               // 16 floats scratch

    // difficulty -> out[1..B]
    sspl_rowsumsq<<<B_N, 256, 0, stream>>>(loss, grads, out);

    // curriculum weight constants (CURRENT_EPOCH=10, MAX_EPOCH=50)
    const double W_FIRST = 2.0 - 10.0 * (2.0 / 49.0);
    const double W_LAST  = 2.0 - W_FIRST;
    const double STEP    = (W_FIRST - W_LAST) / (double)(B_N - 1);

    sspl_rank<<<B_N / 256, 256, 0, stream>>>(loss, out + 1, partials,
                                             (float)W_FIRST, (float)STEP);
    sspl_final<<<1, 32, 0, stream>>>(partials, out);
}